// AugmentPipe_69106023793156
// MI455X (gfx1250) — compile-verified
//
#include <hip/hip_runtime.h>
#include <math.h>

// ---------------------------------------------------------------------------
// AugmentPipe for MI455X (gfx1250): memory-bound fused augmentation.
//   Pass A: params (per-image scalars)
//   Pass B: fused rotate∘flip∘crop (16-tap composed bilinear) + delta + mag1,
//           per-block deterministic partial sums
//   Pass C: deterministic per-image reduction -> mean
//   Pass D: in-place (v-mean)*mag2+mean, streamed through LDS with
//           global_load_async_to_lds_b128 double buffering (CDNA5 async path)
// NOTE: aug_final is defined FIRST so the disasm snippet shows the async ops.
// ---------------------------------------------------------------------------

#define BATCH 256
#define CHN   3
#define HH    256
#define WW    256
#define PLANE (HH * WW)          // 65536
#define IMGSZ (CHN * PLANE)      // 196608 floats per image
#define PSTRIDE 12               // floats of params per image

// ws layout in floats
#define WS_PARAMS 0
#define WS_PART   (BATCH * PSTRIDE)           // 65536 block partials
#define WS_MEAN   (WS_PART + BATCH * HH)      // 256 means

#ifndef __has_builtin
#define __has_builtin(x) 0
#endif
#if __has_builtin(__builtin_amdgcn_global_load_async_to_lds_b128)
#define HAVE_ASYNC_LDS 1
#else
#define HAVE_ASYNC_LDS 0
#endif

#define AS1 __attribute__((address_space(1)))
#define AS3 __attribute__((address_space(3)))
typedef int v4i __attribute__((vector_size(16)));

// ---------------------------------------------------------------------------
// Pass D: in-place (v - mean) * mag2 + mean, streamed through LDS with
// double-buffered async global->LDS B128 loads (CDNA5 async-tensor path).
// ---------------------------------------------------------------------------
#define FBPI  24   // blocks per image
#define FITER 8    // chunks per block; FBPI*FITER*256 float4 == IMGSZ/4

__global__ void __launch_bounds__(256)
aug_final(float* __restrict__ buf, const float* __restrict__ mean,
          const float* __restrict__ P) {
  int b = blockIdx.x / FBPI;
  int blk = blockIdx.x % FBPI;
  float mg2 = P[b * PSTRIDE + 9];
  float mu = mean[b];
  size_t base = (size_t)b * IMGSZ + (size_t)blk * (FITER * 256 * 4);
  float4* g = (float4*)(buf + base);
  int t = threadIdx.x;

#if HAVE_ASYNC_LDS
  __shared__ float4 sbuf[2][256];
  // prime the pipe: chunk 0
  __builtin_amdgcn_global_load_async_to_lds_b128(
      (AS1 v4i*)(g + t),
      (AS3 v4i*)(&sbuf[0][t]), 0, 0);
#pragma unroll
  for (int it = 0; it < FITER; ++it) {
    if (it + 1 < FITER) {
      __builtin_amdgcn_global_load_async_to_lds_b128(
          (AS1 v4i*)(g + (size_t)(it + 1) * 256 + t),
          (AS3 v4i*)(&sbuf[(it + 1) & 1][t]), 0, 0);
      asm volatile("s_wait_asynccnt 1" ::: "memory");
    } else {
      asm volatile("s_wait_asynccnt 0" ::: "memory");
    }
    float4 v = sbuf[it & 1][t];
    v.x = (v.x - mu) * mg2 + mu;
    v.y = (v.y - mu) * mg2 + mu;
    v.z = (v.z - mu) * mg2 + mu;
    v.w = (v.w - mu) * mg2 + mu;
    g[(size_t)it * 256 + t] = v;
  }
#else
#pragma unroll
  for (int it = 0; it < FITER; ++it) {
    float4 v = g[(size_t)it * 256 + t];
    v.x = (v.x - mu) * mg2 + mu;
    v.y = (v.y - mu) * mg2 + mu;
    v.z = (v.z - mu) * mg2 + mu;
    v.w = (v.w - mu) * mg2 + mu;
    g[(size_t)it * 256 + t] = v;
  }
#endif
}

// ---------------------------------------------------------------------------
// Pass A: per-image parameters
// ---------------------------------------------------------------------------
__global__ void aug_params(const float* __restrict__ th, const float* __restrict__ fl,
                           const float* __restrict__ sz, const float* __restrict__ sh,
                           const float* __restrict__ de, const float* __restrict__ m1,
                           const float* __restrict__ m2, float* __restrict__ P) {
  int b = blockIdx.x * blockDim.x + threadIdx.x;
  if (b >= BATCH) return;
  const float PI = 3.14159265358979323846f;
  float theta = (th[b] * 2.0f - 1.0f) * PI;            // ROT_DEG = 180
  float c = cosf(theta), s = sinf(theta);
  float flip = (fl[b] > 0.5f) ? 1.0f : 0.0f;
  // jnp.round == round-half-even -> rintf
  float size_y = rintf((float)HH * (sz[b * 2 + 0] * 0.25f + 1.0f) - 0.5f);
  float size_x = rintf((float)WW * (sz[b * 2 + 1] * 0.25f + 1.0f) - 0.5f);
  float max_y = size_y - 256.0f;                       // CROP = 256
  float max_x = size_x - 256.0f;
  float rng_y = max_y - 1e-5f;
  float rng_x = max_x - 1e-5f;
  float shift_y = rintf(sh[b * 2 + 0] * rng_y - rng_y * 0.5f);
  float shift_x = rintf(sh[b * 2 + 1] * rng_x - rng_x * 0.5f);
  float start_y = floorf(max_y * 0.5f) + shift_y;
  float start_x = floorf(max_x * 0.5f) + shift_x;
  float scale_y = (float)(HH - 1) / (size_y - 1.0f);
  float scale_x = (float)(WW - 1) / (size_x - 1.0f);
  float delta = (de[b] * 2.0f - 1.0f) * 0.3f;          // COLOR = 0.3
  float mag1 = (m1[b] * 2.0f - 1.0f) * 0.3f + 1.0f;
  float mag2 = (m2[b] * 2.0f - 1.0f) * 0.3f + 1.0f;
  float* p = P + b * PSTRIDE;
  p[0] = c;       p[1] = s;       p[2] = flip;
  p[3] = start_y; p[4] = scale_y; p[5] = start_x; p[6] = scale_x;
  p[7] = delta;   p[8] = mag1;    p[9] = mag2;
  p[10] = 0.0f;   p[11] = 0.0f;
}

// ---------------------------------------------------------------------------
// Rotation sample (zero-padded bilinear) of the flipped+rotated image at
// integer coords (yy, xx); accumulates all 3 channels with weight w.
// Coordinates are channel-independent, so computed once for 3 loads per tap.
// ---------------------------------------------------------------------------
__device__ __forceinline__ void rot_sample(const float* __restrict__ imgb,
                                           float cth, float sth, bool flip,
                                           int yy, int xx, float w, float acc[3]) {
  int xs = flip ? (WW - 1 - xx) : xx;
  float xg = (2.0f * (float)xs + 1.0f) * (1.0f / (float)WW) - 1.0f;
  float yg = (2.0f * (float)yy + 1.0f) * (1.0f / (float)HH) - 1.0f;
  float xp = cth * xg - sth * yg;
  float yp = sth * xg + cth * yg;
  float ix = ((xp + 1.0f) * (float)WW - 1.0f) * 0.5f;
  float iy = ((yp + 1.0f) * (float)HH - 1.0f) * 0.5f;
  float fx0 = floorf(ix), fy0 = floorf(iy);
  float bx = ix - fx0, by = iy - fy0;
#pragma unroll
  for (int dy = 0; dy < 2; ++dy) {
    float yt = fy0 + (float)dy;
    float wy = dy ? by : (1.0f - by);
    bool vy = (yt >= 0.0f) && (yt <= (float)(HH - 1));
    int yc = min(max((int)yt, 0), HH - 1);
#pragma unroll
    for (int dx = 0; dx < 2; ++dx) {
      float xt = fx0 + (float)dx;
      float wx = dx ? bx : (1.0f - bx);
      bool v = vy && (xt >= 0.0f) && (xt <= (float)(WW - 1));
      int xc = min(max((int)xt, 0), WW - 1);
      float tw = w * wy * wx;
      if (v) {
        int off = yc * WW + xc;
        acc[0] += tw * imgb[off];
        acc[1] += tw * imgb[PLANE + off];
        acc[2] += tw * imgb[2 * PLANE + off];
      }
    }
  }
}

// ---------------------------------------------------------------------------
// Pass B: one block per output row of one image. 256 threads (8 wave32).
// ---------------------------------------------------------------------------
__global__ void __launch_bounds__(256)
aug_main(const float* __restrict__ img, const float* __restrict__ P,
         float* __restrict__ out, float* __restrict__ part) {
  int b = blockIdx.x >> 8;
  int y = blockIdx.x & 255;
  int x = threadIdx.x;
  const float* p = P + b * PSTRIDE;
  float cth = p[0], sth = p[1];
  bool flip = p[2] > 0.5f;

  // crop/resize coordinates (clamped bilinear, no zero pad)
  float sy = fminf(fmaxf((p[3] + (float)y) * p[4], 0.0f), (float)(HH - 1));
  float sx = fminf(fmaxf((p[5] + (float)x) * p[6], 0.0f), (float)(WW - 1));
  float fy0 = floorf(sy), fx0 = floorf(sx);
  float wy = sy - fy0, wx = sx - fx0;
  int y0 = (int)fy0, x0 = (int)fx0;
  int y1 = min(y0 + 1, HH - 1), x1 = min(x0 + 1, WW - 1);

  const float* imgb = img + (size_t)b * IMGSZ;
  float acc[3] = {0.0f, 0.0f, 0.0f};
  rot_sample(imgb, cth, sth, flip, y0, x0, (1.0f - wy) * (1.0f - wx), acc);
  rot_sample(imgb, cth, sth, flip, y0, x1, (1.0f - wy) * wx, acc);
  rot_sample(imgb, cth, sth, flip, y1, x0, wy * (1.0f - wx), acc);
  rot_sample(imgb, cth, sth, flip, y1, x1, wy * wx, acc);

  float delta = p[7], mag1 = p[8];
  float g0 = acc[0] + delta, g1 = acc[1] + delta, g2 = acc[2] + delta;
  float m = (g0 + g1 + g2) * (1.0f / 3.0f);
  float h0 = (g0 - m) * mag1 + m;
  float h1 = (g1 - m) * mag1 + m;
  float h2 = (g2 - m) * mag1 + m;

  size_t o = (size_t)b * IMGSZ + (size_t)y * WW + (size_t)x;
  out[o] = h0;
  out[o + PLANE] = h1;
  out[o + 2 * PLANE] = h2;

  // deterministic block reduction (fixed tree order)
  __shared__ float red[256];
  red[threadIdx.x] = h0 + h1 + h2;
  __syncthreads();
#pragma unroll
  for (int s = 128; s > 0; s >>= 1) {
    if (threadIdx.x < s) red[threadIdx.x] += red[threadIdx.x + s];
    __syncthreads();
  }
  if (threadIdx.x == 0) part[blockIdx.x] = red[0];
}

// ---------------------------------------------------------------------------
// Pass C: per-image mean (deterministic tree)
// ---------------------------------------------------------------------------
__global__ void __launch_bounds__(256)
aug_reduce(const float* __restrict__ part, float* __restrict__ mean) {
  int b = blockIdx.x;
  __shared__ float red[256];
  red[threadIdx.x] = part[b * HH + threadIdx.x];
  __syncthreads();
#pragma unroll
  for (int s = 128; s > 0; s >>= 1) {
    if (threadIdx.x < s) red[threadIdx.x] += red[threadIdx.x + s];
    __syncthreads();
  }
  if (threadIdx.x == 0) mean[b] = red[0] * (1.0f / (float)IMGSZ);
}

// ---------------------------------------------------------------------------
extern "C" void kernel_launch(void* const* d_in, const int* in_sizes, int n_in,
                              void* d_out, int out_size, void* d_ws, size_t ws_size,
                              hipStream_t stream) {
  const float* img = (const float*)d_in[0];
  const float* th  = (const float*)d_in[1];
  const float* fl  = (const float*)d_in[2];
  const float* sz  = (const float*)d_in[3];
  const float* sh  = (const float*)d_in[4];
  const float* de  = (const float*)d_in[5];
  const float* m1  = (const float*)d_in[6];
  const float* m2  = (const float*)d_in[7];
  float* out = (float*)d_out;
  float* ws  = (float*)d_ws;
  float* P    = ws + WS_PARAMS;
  float* part = ws + WS_PART;
  float* mean = ws + WS_MEAN;

  aug_params<<<1, 256, 0, stream>>>(th, fl, sz, sh, de, m1, m2, P);
  aug_main<<<BATCH * HH, 256, 0, stream>>>(img, P, out, part);
  aug_reduce<<<BATCH, 256, 0, stream>>>(part, mean);
  aug_final<<<BATCH * FBPI, 256, 0, stream>>>(out, mean, P);
}